// WKSLayer_13348758356483
// MI455X (gfx1250) — compile-verified
//
#include <hip/hip_runtime.h>
#include <math.h>

#define BATCH 64
#define LEN   8192
#define FEAT  64
#define WIN   20
#define STP   10
#define NW    818                     // (LEN-WIN)/STP + 1
#define HALF  (NW * FEAT)             // 52352
#define PERB  (2 * HALF)              // 104704
#define CHUNK 16                      // windows per block
#define NCHUNK ((NW + CHUNK - 1) / CHUNK)   // 52
#define SAMP  ((CHUNK - 1) * STP + WIN)     // 170 samples per tile
#define TILE_F (SAMP * FEAT)                // 10880 floats = 43520 B

// Normalized weights, precomputed at compile time.
// EW[k] = exp(-1 + k/19) / sum  ==  r^k / sum(r^k), r = e^(1/19); sums to 1.0 (checked: 1.00000002)
__constant__ constexpr float EW[WIN] = {
    0.02897372f, 0.03053950f, 0.03218986f, 0.03392947f, 0.03576306f,
    0.03769577f, 0.03973289f, 0.04188009f, 0.04414336f, 0.04652890f,
    0.04904336f, 0.05169373f, 0.05448732f, 0.05743189f, 0.06053559f,
    0.06380700f, 0.06725519f, 0.07088978f, 0.07472079f, 0.07875875f};
// LW[k] = (21+k)/510 for k<10, (40-k)/510 for k>=10; sums to exactly 1.
__constant__ constexpr float LW[WIN] = {
    0.04117647f, 0.04313725f, 0.04509804f, 0.04705882f, 0.04901961f,
    0.05098039f, 0.05294118f, 0.05490196f, 0.05686275f, 0.05882353f,
    0.05882353f, 0.05686275f, 0.05490196f, 0.05294118f, 0.05098039f,
    0.04901961f, 0.04705882f, 0.04509804f, 0.04313725f, 0.04117647f};

#if defined(__HIP_DEVICE_COMPILE__) && __has_builtin(__builtin_amdgcn_global_load_async_to_lds_b128)
#define USE_ASYNC 1
#else
#define USE_ASYNC 0
#endif

#if USE_ASYNC
typedef int v4i __attribute__((vector_size(16)));
typedef __attribute__((address_space(1))) v4i as1_v4i;
typedef __attribute__((address_space(3))) v4i as3_v4i;
#endif

__device__ __forceinline__ void g2l_b128(const float4* __restrict__ g, float4* l) {
#if USE_ASYNC
    // global -> LDS async DMA, 16 bytes per lane. Value-preserving casts:
    // flat global addr == AS1 addr; low 32 bits of a flat LDS addr == LDS offset.
    as1_v4i* gp = (as1_v4i*)(unsigned long long)(const void*)g;
    as3_v4i* lp = (as3_v4i*)(unsigned int)(unsigned long long)(void*)l;
    __builtin_amdgcn_global_load_async_to_lds_b128(gp, lp, 0, 0);
#else
    *l = *g;
#endif
}

__device__ __forceinline__ void wait_async_all() {
#if USE_ASYNC
#if __has_builtin(__builtin_amdgcn_s_wait_asynccnt)
    __builtin_amdgcn_s_wait_asynccnt(0);
#else
    asm volatile("s_wait_asynccnt 0" ::: "memory");
#endif
#endif
}

__global__ __launch_bounds__(256) void moments_kernel(const float* __restrict__ x,
                                                      float* __restrict__ out) {
    __shared__ float tile[TILE_F];

    const int tid   = threadIdx.x;
    const int chunk = blockIdx.x;
    const int b     = blockIdx.y;
    const int t0    = chunk * (CHUNK * STP);
    const int nSamp = min(SAMP, LEN - t0);
    const int nVec  = nSamp * (FEAT / 4);   // float4 count to stage

    const float4* __restrict__ src = (const float4*)(x + ((size_t)b * LEN + t0) * FEAT);
    float4* dst = (float4*)tile;
    for (int i = tid; i < nVec; i += 256) {
        g2l_b128(src + i, dst + i);
    }

    wait_async_all();
    __syncthreads();

    const size_t obase = (size_t)b * PERB;
#pragma unroll
    for (int j = 0; j < (CHUNK * FEAT) / 256; ++j) {   // 4 outputs per thread
        const int idx = tid + j * 256;
        const int wl  = idx >> 6;          // local window id
        const int f   = idx & 63;          // feature id
        const int w   = chunk * CHUNK + wl;
        if (w >= NW) continue;

        const float* __restrict__ p = &tile[(wl * STP) * FEAT + f];
        float e1 = 0.f, e2 = 0.f, e3 = 0.f, e4 = 0.f;
        float l1 = 0.f, l2 = 0.f, l3 = 0.f;
#pragma unroll
        for (int k = 0; k < WIN; ++k) {
            const float xv = p[k * FEAT];
            const float ew = EW[k];
            const float lw = LW[k];
            const float x2 = xv * xv;
            const float x3 = x2 * xv;
            e1 = fmaf(ew, xv, e1);
            e2 = fmaf(ew, x2, e2);
            e3 = fmaf(ew, x3, e3);
            e4 = fmaf(ew, x2 * x2, e4);
            l1 = fmaf(lw, xv, l1);
            l2 = fmaf(lw, x2, l2);
            l3 = fmaf(lw, x3, l3);
        }
        // centered moments from raw weighted power sums (sum(w)==1)
        const float mu  = e1, mu2 = mu * mu;
        const float ev  = fmaxf(e2 - mu2, 1e-10f);
        const float m4  = e4 - 4.f * mu * e3 + 6.f * mu2 * e2 - 3.f * mu2 * mu2;
        float kurt = m4 / (ev * ev) - 3.f;
        kurt = fminf(fmaxf(kurt, -100.f), 100.f);

        const float nu  = l1, nu2 = nu * nu;
        const float lv  = fmaxf(l2 - nu2, 1e-10f);
        const float m3  = l3 - 3.f * nu * l2 + 2.f * nu2 * nu;
        const float sq  = sqrtf(lv);
        float skew = m3 / (sq * lv);       // sqrt(lv)^3
        skew = fminf(fmaxf(skew, -100.f), 100.f);

        out[obase + (size_t)w * FEAT + f]        = kurt;
        out[obase + HALF + (size_t)w * FEAT + f] = skew;
    }
}

__global__ __launch_bounds__(256) void stats_kernel(const float* __restrict__ out,
                                                    float* __restrict__ stats) {
    __shared__ float rs[256];
    __shared__ float rq[256];
    const int b = blockIdx.x, tid = threadIdx.x;
    const float* __restrict__ row = out + (size_t)b * PERB;
    float s = 0.f, q = 0.f;
    for (int i = tid; i < PERB; i += 256) {
        const float v = row[i];
        s += v;
        q = fmaf(v, v, q);
    }
    rs[tid] = s; rq[tid] = q;
    __syncthreads();
    for (int off = 128; off > 0; off >>= 1) {
        if (tid < off) { rs[tid] += rs[tid + off]; rq[tid] += rq[tid + off]; }
        __syncthreads();
    }
    if (tid == 0) {
        const float mean = rs[0] * (1.0f / PERB);
        float var = rq[0] * (1.0f / PERB) - mean * mean;
        var = fmaxf(var, 0.f);
        const float sd = sqrtf(var) + 1e-10f;
        stats[b * 2]     = mean;
        stats[b * 2 + 1] = 1.0f / sd;
    }
}

__global__ __launch_bounds__(256) void norm_kernel(float* __restrict__ out,
                                                   const float* __restrict__ stats) {
    const size_t nvec = (size_t)BATCH * PERB / 4;   // PERB % 4 == 0
    float4* o4 = (float4*)out;
    for (size_t i = (size_t)blockIdx.x * blockDim.x + threadIdx.x; i < nvec;
         i += (size_t)gridDim.x * blockDim.x) {
        const int b = (int)((i * 4) / PERB);
        const float mu  = stats[b * 2];
        const float inv = stats[b * 2 + 1];
        float4 v = o4[i];
        v.x = (v.x - mu) * inv;
        v.y = (v.y - mu) * inv;
        v.z = (v.z - mu) * inv;
        v.w = (v.w - mu) * inv;
        o4[i] = v;
    }
}

extern "C" void kernel_launch(void* const* d_in, const int* in_sizes, int n_in,
                              void* d_out, int out_size, void* d_ws, size_t ws_size,
                              hipStream_t stream) {
    (void)in_sizes; (void)n_in; (void)out_size; (void)ws_size;
    const float* x = (const float*)d_in[0];
    float* out   = (float*)d_out;
    float* stats = (float*)d_ws;   // 128 floats

    moments_kernel<<<dim3(NCHUNK, BATCH), 256, 0, stream>>>(x, out);
    stats_kernel<<<BATCH, 256, 0, stream>>>(out, stats);
    norm_kernel<<<2048, 256, 0, stream>>>(out, stats);
}